// decoder_1597727834448
// MI455X (gfx1250) — compile-verified
//
#include <hip/hip_runtime.h>
#include <cmath>

typedef float v2f __attribute__((ext_vector_type(2)));
typedef float v8f __attribute__((ext_vector_type(8)));
typedef int   vi4 __attribute__((vector_size(16)));   // 4 x i32, as expected by async-LDS builtin

#define BB 256
#define TT 127
#define EE 512
#define DD 512

#if defined(__AMDGCN__) && __has_builtin(__builtin_amdgcn_global_load_async_to_lds_b128)
#define HAVE_ASYNC_LDS 1
typedef __attribute__((address_space(1))) vi4* gvi4p;  // global
typedef __attribute__((address_space(3))) vi4* lvi4p;  // LDS
#else
#define HAVE_ASYNC_LDS 0
#endif

__device__ __forceinline__ float sigmoidf_(float x) { return 1.0f / (1.0f + expf(-x)); }

__device__ __forceinline__ void wait_asynccnt0()
{
#if defined(__AMDGCN__)
#if __has_builtin(__builtin_amdgcn_s_wait_asynccnt)
    __builtin_amdgcn_s_wait_asynccnt(0);
#else
    asm volatile("s_wait_asynccnt 0" ::: "memory");
#endif
#endif
}

// ---------------------------------------------------------------------------
// Generic f32 WMMA GEMM: C[M,N] = A[M,K] @ W[K,N] (+ bias[n]) (+ r1v[m]*r1w[n])
// Uses V_WMMA_F32_16X16X4_F32. Layouts per CDNA5 ISA 7.12.2:
//   A 16x4 : lane L holds row M=L%16; vgpr0 = K = (L<16?0:2), vgpr1 = K+1
//   B 4x16 : lane L holds col N=L%16; vgpr0 = K = (L<16?0:2), vgpr1 = K+1
//   C 16x16: vgpr r, lane L -> M = r + (L<16?0:8), N = L%16
// grid: (N/128, M/16); block = 256 threads = 8 waves, wave w covers 16 cols.
// A tile (16 x 512, 32 KB) staged into LDS via GLOBAL_LOAD_ASYNC_TO_LDS_B128
// (ASYNCcnt-tracked, no VGPR round-trip); K consumed in 512-wide chunks.
// ---------------------------------------------------------------------------
__global__ void __launch_bounds__(256)
wmma_gemm_f32(const float* __restrict__ A, const float* __restrict__ W,
              float* __restrict__ C, const float* __restrict__ bias,
              const float* __restrict__ r1v, const float* __restrict__ r1w,
              int M, int N, int K)
{
    __shared__ float As[16 * 512];
    const int m0    = blockIdx.y << 4;
    const int n0    = blockIdx.x << 7;
    const int wave  = threadIdx.x >> 5;
    const int lane  = threadIdx.x & 31;
    const int col   = n0 + (wave << 4) + (lane & 15);
    const int arow  = lane & 15;
    const int khalf = (lane < 16) ? 0 : 2;

    v8f acc = {0.f, 0.f, 0.f, 0.f, 0.f, 0.f, 0.f, 0.f};

    for (int kb = 0; kb < K; kb += 512) {
#if HAVE_ASYNC_LDS
        {   // async DMA of the 16x512 A tile straight into LDS (b128 per lane)
            const float* src = A + (size_t)m0 * K + kb;
            for (int i = threadIdx.x; i < (16 * 512) / 4; i += 256) {
                const int r  = i >> 7;          // float4 index -> row
                const int kk = (i & 127) << 2;  // float4 index -> col
                __builtin_amdgcn_global_load_async_to_lds_b128(
                    (gvi4p)(src + (size_t)r * K + kk),
                    (lvi4p)(As + (i << 2)),
                    0, 0);
            }
            wait_asynccnt0();
        }
#else
        for (int i = threadIdx.x; i < 16 * 512; i += 256) {
            const int r = i >> 9, kk = i & 511;
            As[i] = A[(size_t)(m0 + r) * K + kb + kk];
        }
#endif
        __syncthreads();
        const float* Wk = W + (size_t)kb * N + col;
        for (int k = 0; k < 512; k += 4) {
            const int k0 = k + khalf;
            v2f a, b;
            a.x = As[arow * 512 + k0];
            a.y = As[arow * 512 + k0 + 1];
            b.x = Wk[(size_t)k0 * N];
            b.y = Wk[(size_t)(k0 + 1) * N];
            acc = __builtin_amdgcn_wmma_f32_16x16x4_f32(
                false, a, false, b, (short)0, acc, false, false);
        }
        __syncthreads();
    }

    const float badd = bias ? bias[col] : 0.0f;
    const float rw   = r1w ? r1w[col] : 0.0f;
    const int mbase  = m0 + ((lane < 16) ? 0 : 8);
    for (int r = 0; r < 8; ++r) {
        const int row = mbase + r;
        float v = acc[r] + badd;
        if (r1w) v += r1v[row] * rw;
        C[(size_t)row * N + col] = v;
    }
}

// ---------------------------------------------------------------------------
// Fused attention: scores = tanh(enc_proj + hc_proj) . W_a2 ; softmax over T;
// ctx = attn^T enc ; y_tilde = [ctx, y_t] . W_fc. One block per batch row.
// Working set per step (enc_proj + enc = 133 MB) stays L2-resident (192 MB).
// ---------------------------------------------------------------------------
__global__ void __launch_bounds__(256)
attn_ctx(const float* __restrict__ enc_proj, const float* __restrict__ enc,
         const float* __restrict__ hc_proj, const float* __restrict__ W_a2,
         const float* __restrict__ b_a2, const float* __restrict__ W_fc,
         const float* __restrict__ b_fc, const float* __restrict__ y_hist,
         int t, float* __restrict__ ctx_out, float* __restrict__ y_tilde)
{
    __shared__ float hcp[EE], wa2[EE], sc[TT + 1], red[256];
    const int b = blockIdx.x, tid = threadIdx.x;
    const int wave = tid >> 5, lane = tid & 31;

    for (int e = tid; e < EE; e += 256) { hcp[e] = hc_proj[b * EE + e]; wa2[e] = W_a2[e]; }
    __syncthreads();

    // attention logits: one wave per encoder timestep (round-robin)
    for (int tp = wave; tp < TT; tp += 8) {
        const float* ep = enc_proj + ((size_t)b * TT + tp) * EE;
        float s = 0.f;
        for (int e = lane; e < EE; e += 32) s += wa2[e] * tanhf(ep[e] + hcp[e]);
        for (int off = 16; off; off >>= 1) s += __shfl_down(s, off, 32);
        if (lane == 0) sc[tp] = s + b_a2[0];
    }
    __syncthreads();

    // softmax over 127 logits, done by wave 0
    if (wave == 0) {
        float m = -3.0e38f;
        for (int tp = lane; tp < TT; tp += 32) m = fmaxf(m, sc[tp]);
        for (int off = 16; off; off >>= 1) m = fmaxf(m, __shfl_xor(m, off, 32));
        float ssum = 0.f;
        for (int tp = lane; tp < TT; tp += 32) { float ev = expf(sc[tp] - m); sc[tp] = ev; ssum += ev; }
        for (int off = 16; off; off >>= 1) ssum += __shfl_xor(ssum, off, 32);
        const float inv = 1.0f / ssum;
        for (int tp = lane; tp < TT; tp += 32) sc[tp] *= inv;
    }
    __syncthreads();

    // context + partial fc dot (coalesced across threads per timestep)
    float part = 0.f;
    for (int e = tid; e < EE; e += 256) {
        float a = 0.f;
        const float* ecol = enc + (size_t)b * TT * EE + e;
        for (int tp = 0; tp < TT; ++tp) a += sc[tp] * ecol[(size_t)tp * EE];
        ctx_out[b * EE + e] = a;
        part += a * W_fc[e];
    }
    red[tid] = part;
    __syncthreads();
    for (int s = 128; s; s >>= 1) { if (tid < s) red[tid] += red[tid + s]; __syncthreads(); }
    if (tid == 0) y_tilde[b] = red[0] + y_hist[b * TT + t] * W_fc[EE] + b_fc[0];
}

__global__ void lstm_update(const float* __restrict__ gates,
                            float* __restrict__ h, float* __restrict__ c)
{
    const int i = blockIdx.x * 256 + threadIdx.x;     // over B*D
    const int b = i >> 9, j = i & 511;
    const float* g = gates + (size_t)b * 4 * DD;
    const float gi = g[j], gf = g[DD + j], gg = g[2 * DD + j], go = g[3 * DD + j];
    const float cn = sigmoidf_(gf) * c[i] + sigmoidf_(gi) * tanhf(gg);
    c[i] = cn;
    h[i] = sigmoidf_(go) * tanhf(cn);
}

__global__ void __launch_bounds__(256)
final_pred(const float* __restrict__ h, const float* __restrict__ ctx,
           const float* __restrict__ W_ff, const float* __restrict__ b_ff,
           float* __restrict__ out)
{
    __shared__ float red[256];
    const int b = blockIdx.x, tid = threadIdx.x;
    float p = 0.f;
    for (int j = tid; j < DD; j += 256) p += h[b * DD + j] * W_ff[j];
    for (int e = tid; e < EE; e += 256) p += ctx[b * EE + e] * W_ff[DD + e];
    red[tid] = p;
    __syncthreads();
    for (int s = 128; s; s >>= 1) { if (tid < s) red[tid] += red[tid + s]; __syncthreads(); }
    if (tid == 0) out[b] = red[0] + b_ff[0];
}

__global__ void zero_f32(float* p, int n)
{
    const int i = blockIdx.x * 256 + threadIdx.x;
    if (i < n) p[i] = 0.f;
}

__global__ void add_vec(const float* a, const float* b, float* o, int n)
{
    const int i = blockIdx.x * 256 + threadIdx.x;
    if (i < n) o[i] = a[i] + b[i];
}

__global__ void concat_hc(const float* __restrict__ h, const float* __restrict__ c,
                          float* __restrict__ hc)
{
    const int i = blockIdx.x * 256 + threadIdx.x;     // over B*2D
    const int b = i >> 10, k = i & 1023;
    hc[i] = (k < DD) ? h[b * DD + k] : c[b * DD + (k - DD)];
}

extern "C" void kernel_launch(void* const* d_in, const int* in_sizes, int n_in,
                              void* d_out, int out_size, void* d_ws, size_t ws_size,
                              hipStream_t stream)
{
    const float* enc  = (const float*)d_in[0];   // [B,T,E]
    const float* yh   = (const float*)d_in[1];   // [B,T]
    const float* W_a1 = (const float*)d_in[2];   // [2D+E, E]
    const float* b_a1 = (const float*)d_in[3];   // [E]
    const float* W_a2 = (const float*)d_in[4];   // [E,1]
    const float* b_a2 = (const float*)d_in[5];   // [1]
    const float* W_ih = (const float*)d_in[6];   // [1,4D]
    const float* W_hh = (const float*)d_in[7];   // [D,4D]
    const float* b_ih = (const float*)d_in[8];   // [4D]
    const float* b_hh = (const float*)d_in[9];   // [4D]
    const float* W_fc = (const float*)d_in[10];  // [E+1,1]
    const float* b_fc = (const float*)d_in[11];  // [1]
    const float* W_ff = (const float*)d_in[12];  // [D+E,1]
    const float* b_ff = (const float*)d_in[13];  // [1]

    float* ws = (float*)d_ws;
    size_t off = 0;
    float* enc_proj = ws + off; off += (size_t)BB * TT * EE;   // 66.6 MB (L2-resident)
    float* h        = ws + off; off += (size_t)BB * DD;
    float* c        = ws + off; off += (size_t)BB * DD;        // h,c contiguous
    float* hcbuf    = ws + off; off += (size_t)BB * 2 * DD;
    float* hc_proj  = ws + off; off += (size_t)BB * EE;
    float* ctx      = ws + off; off += (size_t)BB * EE;
    float* ytld     = ws + off; off += BB;
    float* gates    = ws + off; off += (size_t)BB * 4 * DD;
    float* bsum     = ws + off; off += 4 * DD;

    // init recurrent state; fold b_ih+b_hh once
    zero_f32<<<(2 * BB * DD) / 256, 256, 0, stream>>>(h, 2 * BB * DD);
    add_vec<<<(4 * DD + 255) / 256, 256, 0, stream>>>(b_ih, b_hh, bsum, 4 * DD);

    // time-invariant part of attention: enc_proj = enc @ W_a1[2D:,:] + b_a1
    {
        dim3 g(EE / 128, (BB * TT) / 16);
        wmma_gemm_f32<<<g, 256, 0, stream>>>(enc, W_a1 + (size_t)2 * DD * EE, enc_proj,
                                             b_a1, nullptr, nullptr, BB * TT, EE, EE);
    }

    for (int t = 0; t < TT; ++t) {
        concat_hc<<<(BB * 2 * DD) / 256, 256, 0, stream>>>(h, c, hcbuf);
        {   // hc_proj = [h,c] @ W_a1[:2D,:]   (bias already folded into enc_proj)
            dim3 g(EE / 128, BB / 16);
            wmma_gemm_f32<<<g, 256, 0, stream>>>(hcbuf, W_a1, hc_proj,
                                                 nullptr, nullptr, nullptr, BB, EE, 2 * DD);
        }
        attn_ctx<<<BB, 256, 0, stream>>>(enc_proj, enc, hc_proj, W_a2, b_a2,
                                         W_fc, b_fc, yh, t, ctx, ytld);
        {   // gates = h @ W_hh + y_tilde * W_ih + (b_ih + b_hh)
            dim3 g((4 * DD) / 128, BB / 16);
            wmma_gemm_f32<<<g, 256, 0, stream>>>(h, W_hh, gates,
                                                 bsum, ytld, W_ih, BB, 4 * DD, DD);
        }
        lstm_update<<<(BB * DD) / 256, 256, 0, stream>>>(gates, h, c);
    }

    final_pred<<<BB, 256, 0, stream>>>(h, ctx, W_ff, b_ff, (float*)d_out);
}